// Regulator_87746181857564
// MI455X (gfx1250) — compile-verified
//
#include <hip/hip_runtime.h>
#include <hip/hip_bf16.h>
#include <math.h>

typedef __attribute__((ext_vector_type(2))) float v2f;
typedef __attribute__((ext_vector_type(8))) float v8f;

#define Bz 4
#define Sz 128
#define Tz 1024
#define Cz 256
#define CONVz 128
#define Kz 3
#define MLPz 16

// fast silu: v_exp_f32 + v_rcp_f32 (~1 ulp) instead of IEEE div sequence
__device__ __forceinline__ float silu_f(float x){
    return x * __builtin_amdgcn_rcpf(1.f + __expf(-x));
}

// ---------------- K0: sequential cumsum of durations (matches jnp.cumsum order)
__global__ void k0_scan(const float* __restrict__ dur,
                        float* __restrict__ cumstart, float* __restrict__ cumend)
{
    int b = threadIdx.x;
    if (b < Bz) {
        float e = 0.f;
        for (int s = 0; s < Sz; ++s) {
            float d = dur[b*Sz + s];
            e += d;
            cumend[b*Sz + s]   = e;
            cumstart[b*Sz + s] = e - d;
        }
    }
}

// ---------------- K1: conv1d(SAME,K=3) + LayerNorm + silu + (h @ w1[0:128,:]) precompute
// one block per (b,s); 128 threads, one per conv output channel o
__global__ __launch_bounds__(128) void k1_conv(
    const float* __restrict__ ctx,
    const float* __restrict__ wcw, const float* __restrict__ wcb,
    const float* __restrict__ wg,  const float* __restrict__ wbe,
    const float* __restrict__ wd1,
    const float* __restrict__ acw, const float* __restrict__ acb,
    const float* __restrict__ ag,  const float* __restrict__ abe,
    const float* __restrict__ ad1,
    float* __restrict__ hw1w, float* __restrict__ hw1a)
{
    int b = blockIdx.x / Sz, s = blockIdx.x % Sz;
    int o = threadIdx.x;
    __shared__ float ctxs[Kz*Cz];
    __shared__ float red[CONVz];
    __shared__ float hbuf[CONVz];

    for (int idx = o; idx < Kz*Cz; idx += CONVz) {
        int k = idx / Cz, i = idx % Cz;
        int ss = s + k - 1;
        ctxs[idx] = (ss >= 0 && ss < Sz) ? ctx[(b*Sz + ss)*Cz + i] : 0.f;
    }
    __syncthreads();

    float accw = wcb[o], acca = acb[o];
    for (int idx = 0; idx < Kz*Cz; ++idx) {
        float x = ctxs[idx];
        accw = fmaf(x, wcw[idx*CONVz + o], accw);
        acca = fmaf(x, acw[idx*CONVz + o], acca);
    }

    float accs[2]       = { accw, acca };
    const float* gs[2]  = { wg, ag };
    const float* bes[2] = { wbe, abe };
    const float* d1s[2] = { wd1, ad1 };
    float* outs[2]      = { hw1w, hw1a };

    for (int p = 0; p < 2; ++p) {
        float a = accs[p];
        red[o] = a; __syncthreads();
        for (int st = CONVz/2; st > 0; st >>= 1) { if (o < st) red[o] += red[o+st]; __syncthreads(); }
        float mean = red[0] * (1.f/CONVz); __syncthreads();
        red[o] = a*a; __syncthreads();
        for (int st = CONVz/2; st > 0; st >>= 1) { if (o < st) red[o] += red[o+st]; __syncthreads(); }
        float var = red[0] * (1.f/CONVz) - mean*mean; __syncthreads();
        float hh = (a - mean) * __builtin_amdgcn_rsqf(var + 1e-3f) * gs[p][o] + bes[p][o];
        hbuf[o] = silu_f(hh); __syncthreads();
        if (o < MLPz) {
            float acc = 0.f;
            for (int oo = 0; oo < CONVz; ++oo)
                acc = fmaf(hbuf[oo], d1s[p][oo*MLPz + o], acc);
            outs[p][(b*Sz + s)*MLPz + o] = acc;
        }
        __syncthreads();
    }
}

// ---------------- K2: per-(b,t) block of 128 threads (4 waves, one s per thread).
// Both 16->16 MLP layers run on fp32 WMMA (one wave = two 16-row s-tiles).
// D-layout: lane n = output channel o, VGPR v = s-row (v + 8*(lane/16)).
__global__ __launch_bounds__(128) void k2_core(
    const float* __restrict__ mask,
    const float* __restrict__ cumstart, const float* __restrict__ cumend,
    const float* __restrict__ hw1w, const float* __restrict__ hw1a,
    const float* __restrict__ wd1, const float* __restrict__ wd1b,
    const float* __restrict__ wd2, const float* __restrict__ wd2b,
    const float* __restrict__ pww, const float* __restrict__ pwb,
    const float* __restrict__ ad1, const float* __restrict__ ad1b,
    const float* __restrict__ ad2, const float* __restrict__ ad2b,
    float* __restrict__ weights, float* __restrict__ auxS)
{
    int bt = blockIdx.x;
    int b = bt / Tz, t = bt % Tz;
    int s = threadIdx.x;
    int lane = threadIdx.x & 31;
    int wv = threadIdx.x >> 5;
    int n  = lane & 15;
    int hf = lane >> 4;

    __shared__ float xbuf[Sz*18];        // x1; reused as transposed x2*pw buffer
    __shared__ float ybuf[Sz*18];        // y1
    __shared__ float red[Sz];            // softmax reductions
    __shared__ float wbuf[Sz];           // softmax weights
    __shared__ float partials[16*16];    // aux partial sums

    float sp = (float)t - cumstart[b*Sz + s];
    float ep = cumend[b*Sz + s] - sp;

    const float* hw = hw1w + (b*Sz + s)*MLPz;
    const float* ha = hw1a + (b*Sz + s)*MLPz;
#pragma unroll
    for (int j = 0; j < MLPz; ++j) {
        xbuf[s*18 + j] = silu_f(hw[j] + sp*wd1[CONVz*MLPz + j] + ep*wd1[(CONVz+1)*MLPz + j] + wd1b[j]);
        ybuf[s*18 + j] = silu_f(ha[j] + sp*ad1[CONVz*MLPz + j] + ep*ad1[(CONVz+1)*MLPz + j] + ad1b[j]);
    }
    __syncthreads();

    float b2n  = wd2b[n];
    float pwn  = pww[n];
    float pwb0 = pwb[0];

    // ---- logit path: x2 = silu(x1 @ w2 + b2); store x2*pw back (transposed) into xbuf.
    // Safe reuse: a wave reads rows of tile tt before writing those same rows (DS in-order per wave),
    // and waves never touch other waves' rows.
#pragma unroll
    for (int tt = 0; tt < 2; ++tt) {
        int base = wv*32 + tt*16;                     // tile s-base; A row m = n
        v8f acc = {};
#pragma unroll
        for (int k = 0; k < MLPz; k += 4) {
            int ka = k + 2*hf;
            v2f av; av.x = xbuf[(base + n)*18 + ka];  av.y = xbuf[(base + n)*18 + ka + 1];
            v2f bv; bv.x = wd2[ka*MLPz + n];          bv.y = wd2[(ka+1)*MLPz + n];
            acc = __builtin_amdgcn_wmma_f32_16x16x4_f32(false, av, false, bv,
                                                        (short)0, acc, false, false);
        }
#pragma unroll
        for (int v = 0; v < 8; ++v)                   // row m = v + 8*hf, col o = n
            xbuf[(base + v + 8*hf)*18 + n] = silu_f(acc[v] + b2n) * pwn;
    }
    __syncthreads();

    // ---- per-s: logit = sum_o x2pw[s][o] + pwb; then softmax over s
    float logit = pwb0;
#pragma unroll
    for (int o2 = 0; o2 < MLPz; ++o2) logit += xbuf[s*18 + o2];

    float m0 = mask[b*Tz*Sz + s];                     // mask[b, 0, s]
    float wl = m0*logit + (1.f - m0)*(-1e6f);
    red[s] = wl; __syncthreads();
    for (int st = Sz/2; st > 0; st >>= 1) { if (s < st) red[s] = fmaxf(red[s], red[s+st]); __syncthreads(); }
    float mx = red[0]; __syncthreads();
    float e = __expf(wl - mx);
    red[s] = e; __syncthreads();
    for (int st = Sz/2; st > 0; st >>= 1) { if (s < st) red[s] += red[s+st]; __syncthreads(); }
    float inv = __builtin_amdgcn_rcpf(red[0]);
    float wgt = e * inv * mask[(b*Tz + t)*Sz + s];
    weights[(b*Tz + t)*Sz + s] = wgt;
    wbuf[s] = wgt;
    __syncthreads();

    // ---- aux path: y2 = silu(y1 @ aw2 + ab2); auxsum[o] = sum_s y2[s][o]*wgt[s]
    float a2n = ad2b[n];
#pragma unroll
    for (int tt = 0; tt < 2; ++tt) {
        int base = wv*32 + tt*16;
        v8f acc = {};
#pragma unroll
        for (int k = 0; k < MLPz; k += 4) {
            int ka = k + 2*hf;
            v2f av; av.x = ybuf[(base + n)*18 + ka];  av.y = ybuf[(base + n)*18 + ka + 1];
            v2f bv; bv.x = ad2[ka*MLPz + n];          bv.y = ad2[(ka+1)*MLPz + n];
            acc = __builtin_amdgcn_wmma_f32_16x16x4_f32(false, av, false, bv,
                                                        (short)0, acc, false, false);
        }
        float part = 0.f;
#pragma unroll
        for (int v = 0; v < 8; ++v)                    // in-lane reduce over s-rows
            part = fmaf(silu_f(acc[v] + a2n), wbuf[base + v + 8*hf], part);
        partials[((wv*2 + tt)*2 + hf)*16 + n] = part;  // 16 groups x 16 channels
    }
    __syncthreads();
    if (s < MLPz) {
        float accm = 0.f;
        for (int g = 0; g < 16; ++g) accm += partials[g*16 + s];
        auxS[(b*Tz + t)*MLPz + s] = accm * mask[(b*Tz + t)*Sz];   // pre-scale by mask[b,t,0]
    }
}

// ---------------- K3: aligned = weights @ context + auxS @ pa_w + pa_b*mask[b,t,0]
// fp32 WMMA (V_WMMA_F32_16X16X4_F32), one wave per 16x16 output tile
__global__ __launch_bounds__(256) void k3_gemm(
    const float* __restrict__ weights, const float* __restrict__ ctx,
    const float* __restrict__ auxS,    const float* __restrict__ paw,
    const float* __restrict__ pab,     const float* __restrict__ mask,
    float* __restrict__ aligned)
{
    int lane = threadIdx.x & 31;
    int wave = threadIdx.x >> 5;
    int tile = blockIdx.x * 8 + wave;          // 4096 tiles total
    int b   = tile >> 10;                      // 64*16 tiles per batch
    int rem = tile & 1023;
    int tt  = rem >> 4;                        // t-tile 0..63
    int cc  = rem & 15;                        // c-tile 0..15
    int n   = lane & 15;
    int hf  = lane >> 4;
    int col = cc*16 + n;
    int t0  = tt*16;

    v8f acc = {};

    // weights(16x128) @ context(128x16), K=128 in steps of 4
    const float* Wrow = weights + (size_t)(b*Tz + t0 + n)*Sz;   // A row m = lane%16
    const float* Cb   = ctx + (size_t)b*Sz*Cz;
    for (int k = 0; k < Sz; k += 4) {
        int ka = k + 2*hf;
        v2f av = *(const v2f*)(Wrow + ka);                      // A[m][ka], A[m][ka+1]
        v2f bv;
        bv.x = Cb[(size_t)ka*Cz + col];                         // B[ka][n]
        bv.y = Cb[(size_t)(ka+1)*Cz + col];
        acc = __builtin_amdgcn_wmma_f32_16x16x4_f32(false, av, false, bv,
                                                    (short)0, acc, false, false);
    }

    // auxS(16x16) @ pa_w(16x16 slice), K=16, same accumulator
    const float* Arow = auxS + (size_t)(b*Tz + t0 + n)*MLPz;
    for (int k = 0; k < MLPz; k += 4) {
        int ka = k + 2*hf;
        v2f av = *(const v2f*)(Arow + ka);
        v2f bv;
        bv.x = paw[(size_t)ka*Cz + col];
        bv.y = paw[(size_t)(ka+1)*Cz + col];
        acc = __builtin_amdgcn_wmma_f32_16x16x4_f32(false, av, false, bv,
                                                    (short)0, acc, false, false);
    }

    float pb = pab[col];
#pragma unroll
    for (int v = 0; v < 8; ++v) {
        int t = t0 + v + 8*hf;                                  // D row = v + 8*(lane/16)
        float mm = mask[(size_t)(b*Tz + t)*Sz];                 // mask[b,t,0]
        aligned[(size_t)(b*Tz + t)*Cz + col] = acc[v] + pb*mm;
    }
}

extern "C" void kernel_launch(void* const* d_in, const int* in_sizes, int n_in,
                              void* d_out, int out_size, void* d_ws, size_t ws_size,
                              hipStream_t stream)
{
    const float* context  = (const float*)d_in[0];
    const float* durations= (const float*)d_in[1];
    const float* mask     = (const float*)d_in[2];
    const float* w_conv_w = (const float*)d_in[3];
    const float* w_conv_b = (const float*)d_in[4];
    const float* w_ln_g   = (const float*)d_in[5];
    const float* w_ln_b   = (const float*)d_in[6];
    const float* w_d1_w   = (const float*)d_in[7];
    const float* w_d1_b   = (const float*)d_in[8];
    const float* w_d2_w   = (const float*)d_in[9];
    const float* w_d2_b   = (const float*)d_in[10];
    const float* pw_w     = (const float*)d_in[11];
    const float* pw_b     = (const float*)d_in[12];
    const float* a_conv_w = (const float*)d_in[13];
    const float* a_conv_b = (const float*)d_in[14];
    const float* a_ln_g   = (const float*)d_in[15];
    const float* a_ln_b   = (const float*)d_in[16];
    const float* a_d1_w   = (const float*)d_in[17];
    const float* a_d1_b   = (const float*)d_in[18];
    const float* a_d2_w   = (const float*)d_in[19];
    const float* a_d2_b   = (const float*)d_in[20];
    const float* pa_w     = (const float*)d_in[21];
    const float* pa_b     = (const float*)d_in[22];

    float* ws       = (float*)d_ws;
    float* cumstart = ws;                       // 512
    float* cumend   = ws + 512;                 // 512
    float* hw1w     = ws + 1024;                // B*S*16 = 8192
    float* hw1a     = ws + 1024 + 8192;         // 8192
    float* auxS     = ws + 1024 + 16384;        // B*T*16 = 65536

    float* weights = (float*)d_out;             // (B,T,S)
    float* aligned = weights + Bz*Tz*Sz;        // (B,T,C)

    k0_scan<<<1, 32, 0, stream>>>(durations, cumstart, cumend);
    k1_conv<<<Bz*Sz, CONVz, 0, stream>>>(context,
                                         w_conv_w, w_conv_b, w_ln_g, w_ln_b, w_d1_w,
                                         a_conv_w, a_conv_b, a_ln_g, a_ln_b, a_d1_w,
                                         hw1w, hw1a);
    k2_core<<<Bz*Tz, Sz, 0, stream>>>(mask, cumstart, cumend, hw1w, hw1a,
                                      w_d1_w, w_d1_b, w_d2_w, w_d2_b, pw_w, pw_b,
                                      a_d1_w, a_d1_b, a_d2_w, a_d2_b,
                                      weights, auxS);
    int tiles = Bz * (Tz/16) * (Cz/16);         // 4096
    k3_gemm<<<tiles/8, 256, 0, stream>>>(weights, context, auxS, pa_w, pa_b, mask, aligned);
}